// ConservativeLayer_23957327577355
// MI455X (gfx1250) — compile-verified
//
#include <hip/hip_runtime.h>

// ---------------------------------------------------------------------------
// Conservative HEALPix restriction, fully fused, MI455X (gfx1250).
//   out5 = x5 + mean6 ; out6 = x6 - mean6 + mean7 ; out7 = x7 - mean7
// Pure streaming: ~1.06 GB traffic -> ~45 us floor at 23.3 TB/s.
// x7 staged through LDS via gfx1250 async-DMA (ASYNCcnt), double buffered;
// all other traffic is non-temporal b128. Address math is division-free:
// per-iteration slab stride (24576) == 2*kN5, so p5 is loop-invariant and
// bvt advances by +2 -> every base index is a 32-bit counter with a
// compile-time delta.
// ---------------------------------------------------------------------------

typedef __attribute__((ext_vector_type(4))) float vfloat4;

namespace {
constexpr int kN5   = 12 * 4 * 4 * 4 * 4 * 4;   // 12288
constexpr int kN6   = kN5 * 4;                  // 49152
constexpr int kN7   = kN6 * 4;                  // 196608
constexpr int kNT   = kN5 + kN6 + kN7;          // 258048
constexpr int kBVT  = 2 * 2 * 4;                // B*V*T = 16
constexpr long long kSlabs = (long long)kBVT * kN5;  // 196608 zoom-5 parents

constexpr int kTPB           = 128;             // 4 wave32s
constexpr int kWavesPerBlock = kTPB / 32;
constexpr int kBlocks        = 1536;
constexpr int kTotalWaves    = kBlocks * kWavesPerBlock;            // 6144
constexpr int kIters         = (int)(kSlabs / ((long long)kTotalWaves * 4)); // 8
static_assert((long long)kTotalWaves * 4 * kIters == kSlabs, "grid must tile slabs exactly");
static_assert(kTotalWaves * 4 == 2 * kN5, "slab stride must equal 2*kN5 (division-free bvt walk)");

// Per-iteration index deltas (vfloat4 units; bvt advances by 2 each iter).
constexpr unsigned kD5 = 2u * kN5 * 8u;
constexpr unsigned kD6 = 2u * kN6 * 8u;
constexpr unsigned kD7 = 2u * kN7 * 8u;
constexpr unsigned kDO = 2u * kNT * 8u;
}

// One async b128 row copy: per lane 16B global -> LDS. INST_OFFSET is added
// to BOTH the global and the LDS address (ISA 15.18), so a single immediate
// walks both streams row by row (row stride = 32 floats = 128 B).
#define ASYNC_ROW(J)                                                        \
  asm volatile("global_load_async_to_lds_b128 %0, %1, off offset:%2"        \
               :: "v"(lbase), "v"(g), "n"((J) * 128) : "memory")

__global__ __launch_bounds__(kTPB)
void cons_fused_kernel(const float* __restrict__ x5,
                       const float* __restrict__ x6,
                       const float* __restrict__ x7,
                       float* __restrict__ out)
{
  // [wave][stage][slab-in-wave][row 0..15][f-quad] : 4*2*4*16*8*16B = 64 KB
  __shared__ vfloat4 buf[kWavesPerBlock][2][4][16][8];

  const int tid  = threadIdx.x;
  const int wid  = tid >> 5;
  const int lane = tid & 31;
  const int siw  = lane >> 3;   // slab within wave (0..3)
  const int fq   = lane & 7;    // float4 column of F=32 (0..7)
  const int wgid = blockIdx.x * kWavesPerBlock + wid;

  const vfloat4* x5v = reinterpret_cast<const vfloat4*>(x5);
  const vfloat4* x6v = reinterpret_cast<const vfloat4*>(x6);
  const vfloat4* x7v = reinterpret_cast<const vfloat4*>(x7);
  vfloat4*       ov  = reinterpret_cast<vfloat4*>(out);

  // ---- division-free slab decomposition at t=0 ----
  const unsigned sbase = (unsigned)(wgid * 4 + siw);       // < 2*kN5
  const unsigned bvt0  = (sbase >= (unsigned)kN5) ? 1u : 0u;
  const unsigned p5    = sbase - bvt0 * (unsigned)kN5;

  // Loop-carried base indices (vfloat4 units, all < 2^32).
  unsigned i5  = sbase * 8u + (unsigned)fq;                               // x5
  unsigned i6  = (bvt0 * (unsigned)kN6 + 4u * p5) * 8u + (unsigned)fq;    // x6
  unsigned oB  = bvt0 * (unsigned)kNT * 8u;                               // out slab base
  unsigned o5  = oB + p5 * 8u + (unsigned)fq;
  unsigned o6  = oB + ((unsigned)kN5 + 4u * p5) * 8u + (unsigned)fq;
  unsigned o7  = oB + ((unsigned)(kN5 + kN6) + 16u * p5) * 8u + (unsigned)fq;
  const vfloat4* g = x7v + (size_t)((bvt0 * (unsigned)kN7 + 16u * p5) * 8u + (unsigned)fq);

  // LDS byte addresses of this lane's two staging buffers (low 32 bits of
  // the generic pointer == LDS offset on AMDGPU).
  const unsigned lb[2] = {
      (unsigned)(unsigned long long)(const void*)&buf[wid][0][siw][0][fq],
      (unsigned)(unsigned long long)(const void*)&buf[wid][1][siw][0][fq]};

  // ---- async stage: pull 16 zoom-7 rows (2 KB/slab) into LDS ----
  auto issue = [&](const vfloat4* gp, unsigned lbase) {
    const vfloat4* gv = gp;
    const vfloat4* &g = gv;  // name used by ASYNC_ROW
    // WAR guard: prior ds_loads from this buffer must retire before the async
    // engine overwrites it (DS and ASYNC streams are mutually unordered).
    asm volatile("s_wait_dscnt 0" ::: "memory");
    ASYNC_ROW(0);  ASYNC_ROW(1);  ASYNC_ROW(2);  ASYNC_ROW(3);
    ASYNC_ROW(4);  ASYNC_ROW(5);  ASYNC_ROW(6);  ASYNC_ROW(7);
    ASYNC_ROW(8);  ASYNC_ROW(9);  ASYNC_ROW(10); ASYNC_ROW(11);
    ASYNC_ROW(12); ASYNC_ROW(13); ASYNC_ROW(14); ASYNC_ROW(15);
  };

  issue(g, lb[0]);
  int stage = 0;

#pragma unroll 1
  for (int t = 0; t < kIters; ++t) {
    if (t + 1 < kIters) {
      g += (size_t)kD7;
      issue(g, lb[stage ^ 1]);
      // 32 outstanding (prev 16 + just-issued 16); in-order completion, so
      // waiting down to 16 means the previous stage's data is in LDS.
      asm volatile("s_wait_asynccnt 16" ::: "memory");
    } else {
      asm volatile("s_wait_asynccnt 0" ::: "memory");
    }

    // ---- compute: all three zoom levels for this slab column ----
    const vfloat4 a5 = __builtin_nontemporal_load(&x5v[i5]);
    vfloat4 a6[4];
#pragma unroll
    for (int i = 0; i < 4; ++i)
      a6[i] = __builtin_nontemporal_load(&x6v[i6 + (unsigned)i * 8u]);

    vfloat4 m7[4];
#pragma unroll
    for (int i = 0; i < 4; ++i) {
      const vfloat4 r0 = buf[wid][stage][siw][4 * i + 0][fq];
      const vfloat4 r1 = buf[wid][stage][siw][4 * i + 1][fq];
      const vfloat4 r2 = buf[wid][stage][siw][4 * i + 2][fq];
      const vfloat4 r3 = buf[wid][stage][siw][4 * i + 3][fq];
      const vfloat4 m  = ((r0 + r1) + (r2 + r3)) * 0.25f;
      m7[i] = m;
      __builtin_nontemporal_store(r0 - m, &ov[o7 + (unsigned)(4 * i + 0) * 8u]);
      __builtin_nontemporal_store(r1 - m, &ov[o7 + (unsigned)(4 * i + 1) * 8u]);
      __builtin_nontemporal_store(r2 - m, &ov[o7 + (unsigned)(4 * i + 2) * 8u]);
      __builtin_nontemporal_store(r3 - m, &ov[o7 + (unsigned)(4 * i + 3) * 8u]);
    }

    const vfloat4 m6 = ((a6[0] + a6[1]) + (a6[2] + a6[3])) * 0.25f;
#pragma unroll
    for (int i = 0; i < 4; ++i)
      __builtin_nontemporal_store((a6[i] - m6) + m7[i], &ov[o6 + (unsigned)i * 8u]);
    __builtin_nontemporal_store(a5 + m6, &ov[o5]);

    // advance loop-carried indices (bvt += 2)
    i5 += kD5; i6 += kD6; o5 += kDO; o6 += kDO; o7 += kDO;
    stage ^= 1;
  }
}

extern "C" void kernel_launch(void* const* d_in, const int* in_sizes, int n_in,
                              void* d_out, int out_size, void* d_ws, size_t ws_size,
                              hipStream_t stream) {
  (void)in_sizes; (void)n_in; (void)out_size; (void)d_ws; (void)ws_size;
  const float* x5 = (const float*)d_in[0];
  const float* x6 = (const float*)d_in[1];
  const float* x7 = (const float*)d_in[2];
  cons_fused_kernel<<<kBlocks, kTPB, 0, stream>>>(x5, x6, x7, (float*)d_out);
}